// Attention_10101763080837
// MI455X (gfx1250) — compile-verified
//
#include <hip/hip_runtime.h>
#include <math.h>

// ---------------------------------------------------------------------------
// CDNA5 (gfx1250) fp32 WMMA attention layer.
//   Q = rope(x@wq), K = rope(x@wk), V = x@wv
//   O = softmax(Q Kt / sqrt(D)) V ;  out = O @ wo
// All matmuls via V_WMMA_F32_16X16X4_F32 (exact fp32).
// GEMM tiles double-buffered in LDS via GLOBAL_LOAD_ASYNC_TO_LDS (ASYNCcnt).
// ---------------------------------------------------------------------------

typedef __attribute__((ext_vector_type(2))) float v2f;
typedef __attribute__((ext_vector_type(8))) float v8f;

#define N_ROWS   2048   // B*S
#define SEQ      1024
#define DIM      2048
#define KVD      512
#define HQ       32
#define HKV      8
#define HD       64
#define BQ       32     // query rows per attention workgroup

// ---------------- async global->LDS helpers (probe-guarded) ----------------
#if defined(__has_builtin)
#  if __has_builtin(__builtin_amdgcn_global_load_async_to_lds_b32) && \
      __has_builtin(__builtin_amdgcn_global_load_async_to_lds_b128)
#    define HAVE_ASYNC_LDS 1
#  else
#    define HAVE_ASYNC_LDS 0
#  endif
#  if __has_builtin(__builtin_amdgcn_s_wait_asynccnt)
#    define HAVE_WAIT_ASYNC 1
#  else
#    define HAVE_WAIT_ASYNC 0
#  endif
#else
#  define HAVE_ASYNC_LDS 0
#  define HAVE_WAIT_ASYNC 0
#endif

#if HAVE_ASYNC_LDS
typedef int vsi4 __attribute__((vector_size(16)));
typedef __attribute__((address_space(1))) int  gas_i32;   // global int*
typedef __attribute__((address_space(3))) int  las_i32;   // LDS int*
typedef __attribute__((address_space(1))) vsi4 gas_v4i;   // global int4*
typedef __attribute__((address_space(3))) vsi4 las_v4i;   // LDS int4*
#endif

static __device__ __forceinline__ void cp_b32(float* lds_dst, const float* gsrc) {
#if HAVE_ASYNC_LDS
  __builtin_amdgcn_global_load_async_to_lds_b32(
      (gas_i32*)gsrc, (las_i32*)lds_dst, 0, 0);
#else
  *lds_dst = *gsrc;
#endif
}

static __device__ __forceinline__ void cp_b128(float* lds_dst, const float* gsrc) {
#if HAVE_ASYNC_LDS
  __builtin_amdgcn_global_load_async_to_lds_b128(
      (gas_v4i*)gsrc, (las_v4i*)lds_dst, 0, 0);
#else
  *(float4*)lds_dst = *(const float4*)gsrc;
#endif
}

static __device__ __forceinline__ void wait_async_all() {
#if HAVE_ASYNC_LDS
#  if HAVE_WAIT_ASYNC
  __builtin_amdgcn_s_wait_asynccnt(0);
#  else
  asm volatile("s_wait_asynccnt 0" ::: "memory");
#  endif
#endif
}

static __device__ __forceinline__ v8f wmma_f32(v2f a, v2f b, v8f c) {
  // V_WMMA_F32_16X16X4_F32 : D = A(16x4) * B(4x16) + C, fp32 exact
  return __builtin_amdgcn_wmma_f32_16x16x4_f32(
      /*neg_a=*/false, a, /*neg_b=*/false, b,
      /*c_mod=*/(short)0, c, /*reuse_a=*/false, /*reuse_b=*/false);
}

// ---------------------------------------------------------------------------
// fp32 GEMM:  C[M,N] = A[M,K] * B[K,N]  (row-major, block tile 128x128)
// 128 threads = 4 waves arranged 2(m) x 2(n); each wave owns a 64x64 subtile
// (4x4 WMMA accumulators -> 16 WMMAs per k-step vs 8 fragment fetches).
// LDS double-buffered; tiles staged with async global->LDS copies.
// A tile kept row-major with 20-dword padded stride so an A fragment is one
// aligned ds_load_b64 (bank-conflict free).
// ---------------------------------------------------------------------------
#define GA_STRIDE 20   // dwords per A-tile row (16 data + 4 pad, 16B aligned)

__global__ __launch_bounds__(128)
void gemm_f32_wmma(const float* __restrict__ A, const float* __restrict__ B,
                   float* __restrict__ C, int M, int N, int K) {
  __shared__ float As[2][128 * GA_STRIDE]; // [m][k] row-major, padded
  __shared__ float Bs[2][16 * 128];        // [k][n]

  const int tid  = threadIdx.x;
  const int lane = tid & 31;
  const int wid  = tid >> 5;
  const int wm   = wid >> 1;            // 0..1 -> 64-row strip
  const int wn   = wid & 1;             // 0..1 -> 64-col strip
  const int m0   = blockIdx.y * 128;
  const int n0   = blockIdx.x * 128;

  v8f acc[4][4] = {};

  // stage tile for k0 = 0 into buffer 0
  {
    #pragma unroll
    for (int i = 0; i < 4; ++i) {              // A: 512 x 16B quads
      int idx = tid + i * 128;
      int m = idx >> 2, kq = (idx & 3) << 2;
      cp_b128(&As[0][m * GA_STRIDE + kq], &A[(size_t)(m0 + m) * K + kq]);
    }
    #pragma unroll
    for (int i = 0; i < 4; ++i) {              // B: 512 x 16B quads
      int idx = tid + i * 128;
      int k = idx >> 5, n4 = (idx & 31) << 2;
      cp_b128(&Bs[0][k * 128 + n4], &B[(size_t)k * N + n0 + n4]);
    }
  }
  wait_async_all();
  __syncthreads();

  for (int k0 = 0; k0 < K; k0 += 16) {
    const int cur = (k0 >> 4) & 1;
    const int nxt = cur ^ 1;

    // prefetch next k-tile while this one is consumed by the matrix pipe
    if (k0 + 16 < K) {
      #pragma unroll
      for (int i = 0; i < 4; ++i) {
        int idx = tid + i * 128;
        int m = idx >> 2, kq = (idx & 3) << 2;
        cp_b128(&As[nxt][m * GA_STRIDE + kq],
                &A[(size_t)(m0 + m) * K + (k0 + 16 + kq)]);
      }
      #pragma unroll
      for (int i = 0; i < 4; ++i) {
        int idx = tid + i * 128;
        int k = idx >> 5, n4 = (idx & 31) << 2;
        cp_b128(&Bs[nxt][k * 128 + n4],
                &B[(size_t)(k0 + 16 + k) * N + n0 + n4]);
      }
    }

    #pragma unroll
    for (int kk = 0; kk < 16; kk += 4) {
      const int kl = kk + ((lane >> 4) << 1);   // lanes 16..31 take K+2,K+3
      v2f afr[4], bfr[4];
      #pragma unroll
      for (int mi = 0; mi < 4; ++mi) {
        int mrow = wm * 64 + mi * 16 + (lane & 15);
        afr[mi] = *(const v2f*)&As[cur][mrow * GA_STRIDE + kl]; // ds_load_b64
      }
      #pragma unroll
      for (int ni = 0; ni < 4; ++ni) {
        int ncol = wn * 64 + ni * 16 + (lane & 15);
        bfr[ni] = (v2f){Bs[cur][kl * 128 + ncol], Bs[cur][(kl + 1) * 128 + ncol]};
      }
      #pragma unroll
      for (int mi = 0; mi < 4; ++mi)
        #pragma unroll
        for (int ni = 0; ni < 4; ++ni)
          acc[mi][ni] = wmma_f32(afr[mi], bfr[ni], acc[mi][ni]);
    }

    wait_async_all();
    __syncthreads();
  }

  // Writeback: VGPR r holds M = (lane>=16 ? 8 : 0)+r, N = lane&15.
  #pragma unroll
  for (int mi = 0; mi < 4; ++mi)
    #pragma unroll
    for (int ni = 0; ni < 4; ++ni)
      #pragma unroll
      for (int r = 0; r < 8; ++r) {
        int row = m0 + wm * 64 + mi * 16 + ((lane >> 4) << 3) + r;
        int col = n0 + wn * 64 + ni * 16 + (lane & 15);
        C[(size_t)row * N + col] = acc[mi][ni][r];
      }
}

// ---------------------------------------------------------------------------
// RoPE in-place on a [rows, H*D] buffer; pair (2i, 2i+1) within each head.
// ---------------------------------------------------------------------------
__global__ __launch_bounds__(256)
void rope_f32(float* __restrict__ buf, int H, int total_pairs) {
  int idx = blockIdx.x * blockDim.x + threadIdx.x;
  if (idx >= total_pairs) return;
  const int half = HD / 2;
  int i   = idx % half;
  int h   = (idx / half) % H;
  int row = idx / (half * H);
  int s   = row % SEQ;
  float inv = __expf(-((float)(2 * i) / (float)HD) * __logf(10000.0f));
  float ang = (float)s * inv;
  float sn, cs;
  sincosf(ang, &sn, &cs);
  float* p = buf + ((size_t)row * H + h) * HD + 2 * i;
  float x1 = p[0], x2 = p[1];
  p[0] = x1 * cs - x2 * sn;
  p[1] = x1 * sn + x2 * cs;
}

// ---------------------------------------------------------------------------
// Attention: one workgroup = (32 queries) x (one q-head) x (one batch).
// Full 32x1024 score panel lives in LDS (CDNA5: 320KB/WGP), padded +1.
// K tile staged transposed (async b32), V tile staged row-major (async b128)
// reusing the same LDS buffer.  8 waves, 256 threads.
// ---------------------------------------------------------------------------
#define VT_STRIDE 68   // dwords per V-tile row (64 data + 4 pad, 16B aligned)

__global__ __launch_bounds__(256)
void attn_f32_wmma(const float* __restrict__ Q, const float* __restrict__ Kb,
                   const float* __restrict__ Vb, float* __restrict__ O) {
  __shared__ float sc[BQ][SEQ + 1];      // 32 x 1025 fp32  (~128 KB)
  __shared__ float qs[HD][BQ + 1];       // Q tile transposed [d][q]
  __shared__ float ktvt[64 * VT_STRIDE]; // K tile [d][key] (stride 65) then
                                         // V tile [key][d] (stride 68)

  const int qb = blockIdx.x;          // 0..31  query block
  const int h  = blockIdx.y;          // 0..31  q head
  const int b  = blockIdx.z;          // 0..1   batch
  const int g  = h >> 2;              // kv head (n_rep = 4)

  const int tid  = threadIdx.x;
  const int lane = tid & 31;
  const int wid  = tid >> 5;
  const int mi   = wid & 1;           // 16-row half of the 32-query block
  const int ni   = wid >> 1;          // 0..3 : 16-col group within 64
  const float scale = 0.125f;         // 1/sqrt(64)

  const size_t qrow0 = (size_t)b * SEQ + (size_t)qb * BQ;

  // ---- stage Q tile (32x64) transposed into LDS ----
  #pragma unroll
  for (int i = 0; i < 8; ++i) {
    int idx = tid + i * 256;
    int d = idx & 63, q = idx >> 6;
    cp_b32(&qs[d][q], &Q[(qrow0 + q) * DIM + h * HD + d]);
  }

  // ---- scores: S = Q * K^T over key chunks of 64 ----
  const int mrow = mi * 16 + (lane & 15);
  const int ncol = ni * 16 + (lane & 15);
  for (int kv0 = 0; kv0 < SEQ; kv0 += 64) {
    __syncthreads();                  // prev chunk consumed
    #pragma unroll
    for (int i = 0; i < 16; ++i) {    // 64x64 K tile, transposed, coalesced on d
      int idx = tid + i * 256;
      int d = idx & 63, ky = idx >> 6;
      cp_b32(&ktvt[d * 65 + ky],
             &Kb[((size_t)b * SEQ + kv0 + ky) * KVD + g * HD + d]);
    }
    wait_async_all();
    __syncthreads();

    v8f acc = {};
    #pragma unroll
    for (int kk = 0; kk < HD; kk += 4) {
      const int kl = kk + ((lane >> 4) << 1);
      v2f a  = (v2f){qs[kl][mrow], qs[kl + 1][mrow]};
      v2f bb = (v2f){ktvt[kl * 65 + ncol], ktvt[(kl + 1) * 65 + ncol]};
      acc = wmma_f32(a, bb, acc);
    }
    #pragma unroll
    for (int r = 0; r < 8; ++r) {
      int row = mi * 16 + ((lane >> 4) << 3) + r;
      int col = kv0 + ni * 16 + (lane & 15);
      sc[row][col] = acc[r];
    }
  }
  __syncthreads();

  // ---- softmax over each of the 32 rows; wave w owns rows 4w..4w+3 ----
  #pragma unroll
  for (int j = 0; j < 4; ++j) {
    const int r = wid * 4 + j;
    float mx = -INFINITY;
    for (int c = lane; c < SEQ; c += 32) mx = fmaxf(mx, sc[r][c]);
    #pragma unroll
    for (int off = 16; off; off >>= 1) mx = fmaxf(mx, __shfl_xor(mx, off, 32));
    float sum = 0.0f;
    for (int c = lane; c < SEQ; c += 32) {
      float p = __expf(scale * (sc[r][c] - mx));
      sc[r][c] = p;
      sum += p;
    }
    #pragma unroll
    for (int off = 16; off; off >>= 1) sum += __shfl_xor(sum, off, 32);
    float rinv = 1.0f / sum;
    for (int c = lane; c < SEQ; c += 32) sc[r][c] *= rinv;
  }

  // ---- O = P(32x1024) @ V(1024x64); V chunks staged through LDS ----
  const float* Vg = Vb + (size_t)b * SEQ * KVD + g * HD;
  v8f oac = {};
  for (int kv0 = 0; kv0 < SEQ; kv0 += 64) {
    __syncthreads();                  // prev V chunk consumed / softmax done
    #pragma unroll
    for (int i = 0; i < 4; ++i) {     // 64x64 V tile, row-major, b128
      int idx = tid + i * 256;
      int ky = idx >> 4, d4 = (idx & 15) << 2;
      cp_b128(&ktvt[ky * VT_STRIDE + d4],
              &Vg[(size_t)(kv0 + ky) * KVD + d4]);
    }
    wait_async_all();
    __syncthreads();

    #pragma unroll
    for (int kk = 0; kk < 64; kk += 4) {
      const int kl = kk + ((lane >> 4) << 1);
      const int kg = kv0 + kl;
      v2f a  = (v2f){sc[mrow][kg], sc[mrow][kg + 1]};
      v2f bb = (v2f){ktvt[kl * VT_STRIDE + ncol],
                     ktvt[(kl + 1) * VT_STRIDE + ncol]};
      oac = wmma_f32(a, bb, oac);
    }
  }

  #pragma unroll
  for (int r = 0; r < 8; ++r) {
    int q = mi * 16 + ((lane >> 4) << 3) + r;
    O[(qrow0 + q) * DIM + h * HD + ncol] = oac[r];
  }
}

// ---------------------------------------------------------------------------
extern "C" void kernel_launch(void* const* d_in, const int* in_sizes, int n_in,
                              void* d_out, int out_size, void* d_ws, size_t ws_size,
                              hipStream_t stream) {
  const float* x  = (const float*)d_in[0];
  const float* wq = (const float*)d_in[1];
  const float* wk = (const float*)d_in[2];
  const float* wv = (const float*)d_in[3];
  const float* wo = (const float*)d_in[4];

  float* ws = (float*)d_ws;
  float* Qb = ws;                              // 2048*2048
  float* Kb = Qb + (size_t)N_ROWS * DIM;       // 2048*512
  float* Vb = Kb + (size_t)N_ROWS * KVD;       // 2048*512
  float* Ab = Vb + (size_t)N_ROWS * KVD;       // 2048*2048 (attention output)

  dim3 gblk(128);
  dim3 ablk(256);

  // QKV projections
  gemm_f32_wmma<<<dim3(DIM / 128, N_ROWS / 128), gblk, 0, stream>>>(x, wq, Qb, N_ROWS, DIM, DIM);
  gemm_f32_wmma<<<dim3(KVD / 128, N_ROWS / 128), gblk, 0, stream>>>(x, wk, Kb, N_ROWS, KVD, DIM);
  gemm_f32_wmma<<<dim3(KVD / 128, N_ROWS / 128), gblk, 0, stream>>>(x, wv, Vb, N_ROWS, KVD, DIM);

  // RoPE on Q and K
  {
    int pq = N_ROWS * HQ * (HD / 2);
    int pk = N_ROWS * HKV * (HD / 2);
    rope_f32<<<(pq + 255) / 256, ablk, 0, stream>>>(Qb, HQ, pq);
    rope_f32<<<(pk + 255) / 256, ablk, 0, stream>>>(Kb, HKV, pk);
  }

  // Attention
  attn_f32_wmma<<<dim3(SEQ / BQ, HQ, 2), ablk, 0, stream>>>(Qb, Kb, Vb, Ab);

  // Output projection
  gemm_f32_wmma<<<dim3(DIM / 128, N_ROWS / 128), gblk, 0, stream>>>(Ab, wo, (float*)d_out, N_ROWS, DIM, DIM);
}